// FeatureAlign_V2_34376918237270
// MI455X (gfx1250) — compile-verified
//
#include <hip/hip_runtime.h>
#include <hip/hip_bf16.h>

// ---------------- problem constants (from reference) ----------------
#define IN_NC  128
#define OUT_NC 128
#define Gg     8
#define CG     16
#define Bc     4
#define Hc     128
#define Wc     128
#define HWc    (Hc * Wc)
#define HSc    64
#define WSc    64
#define OMCH   216   // 144 offset + 72 mask
#define ZCH    256

typedef __attribute__((ext_vector_type(2))) float v2f;
typedef __attribute__((ext_vector_type(8))) float v8f;

__device__ __forceinline__ int imin(int a, int b) { return a < b ? a : b; }
__device__ __forceinline__ int imax(int a, int b) { return a > b ? a : b; }
__device__ __forceinline__ float sigmoidf(float x) { return 1.0f / (1.0f + __expf(-x)); }

// ---------------- kernel 1: per-(b,c) global average pool ----------------
__global__ __launch_bounds__(256) void pool_kernel(const float* __restrict__ feat_l,
                                                   float* __restrict__ pooled) {
    __shared__ float red[256];
    const int bc = blockIdx.x;                 // 0 .. B*128-1
    const float* p = feat_l + (size_t)bc * HWc;
    float s = 0.0f;
    #pragma unroll 4
    for (int i = 0; i < HWc / 256; ++i) s += p[threadIdx.x + i * 256];
    red[threadIdx.x] = s;
    __syncthreads();
    for (int off = 128; off > 0; off >>= 1) {
        if ((int)threadIdx.x < off) red[threadIdx.x] += red[threadIdx.x + off];
        __syncthreads();
    }
    if (threadIdx.x == 0) pooled[bc] = red[0] * (1.0f / (float)HWc);
}

// ---------------- kernel 2: atten = sigmoid(pooled @ W^T) ----------------
__global__ __launch_bounds__(128) void atten_kernel(const float* __restrict__ pooled,
                                                    const float* __restrict__ w,
                                                    float* __restrict__ atten) {
    const int b = blockIdx.x;
    const int o = threadIdx.x;
    float s = 0.0f;
    #pragma unroll 8
    for (int c = 0; c < IN_NC; ++c) s += pooled[b * IN_NC + c] * w[o * IN_NC + c];
    atten[b * IN_NC + o] = sigmoidf(s);
}

// ---------------- kernel 3: bilinear 2x upsample, group-major ch-innermost ----------------
// fup[((b*G + g)*HW + y*W + x)*16 + cg] = bilinear(feat_s[b, g*16+cg], half-pixel)
__global__ __launch_bounds__(256) void upsample_kernel(const float* __restrict__ fs,
                                                       float* __restrict__ fup) {
    int n = blockIdx.x * 256 + threadIdx.x;    // Bc*Gg*HWc*CG elements
    int cg = n & 15;
    int r = n >> 4;
    int x = r & (Wc - 1); r >>= 7;
    int y = r & (Hc - 1); r >>= 7;
    int g = r & (Gg - 1);
    int b = r >> 3;
    float sy = ((float)y + 0.5f) * 0.5f - 0.5f;
    float sx = ((float)x + 0.5f) * 0.5f - 0.5f;
    float fy = floorf(sy), fx = floorf(sx);
    float wy = sy - fy, wx = sx - fx;
    int y0 = (int)fy, x0 = (int)fx;
    int y0c = imax(y0, 0), y1c = imin(y0 + 1, HSc - 1);
    int x0c = imax(x0, 0), x1c = imin(x0 + 1, WSc - 1);
    const float* base = fs + (size_t)(b * OUT_NC + g * CG + cg) * (HSc * WSc);
    float v00 = base[y0c * WSc + x0c];
    float v01 = base[y0c * WSc + x1c];
    float v10 = base[y1c * WSc + x0c];
    float v11 = base[y1c * WSc + x1c];
    fup[n] = (1.0f - wy) * ((1.0f - wx) * v00 + wx * v01)
           +         wy  * ((1.0f - wx) * v10 + wx * v11);
}

// ---------------- kernel 4: fused fsm-conv + om-conv + DCNv2 per 16-pixel tile ----------------
#define XS_S 130   // xs row stride (16 x 128 A tile)
#define ZS_S 258   // zs row stride (16 x 256: [feat_arm | 2*feat_up])
#define OM_S 220   // om row stride (16 x 216 padded)

__global__ __launch_bounds__(256) void mega_kernel(
    const float* __restrict__ feat_l, const float* __restrict__ atten,
    const float* __restrict__ fup,    const float* __restrict__ fsm_conv_w,
    const float* __restrict__ om_w,   const float* __restrict__ om_b,
    const float* __restrict__ dcn_w,  const float* __restrict__ dcn_b,
    float* __restrict__ out)
{
    __shared__ float xs[16 * XS_S];
    __shared__ float zs[16 * ZS_S];
    __shared__ float oms[16 * OM_S];

    const int tid  = threadIdx.x;
    const int lane = tid & 31;
    const int wave = tid >> 5;
    const int p0   = blockIdx.x * 16;          // 16 consecutive pixels: same b, same y
    const int b    = p0 >> 14;                 // / (128*128)
    const int rem  = p0 & (HWc - 1);
    const int ypos = rem >> 7;
    const int x0   = rem & (Wc - 1);

    // ---- stage A tile for GEMM1: xs[m][c] = feat_l * (1 + atten) ----
    #pragma unroll
    for (int i = 0; i < 8; ++i) {
        int idx = i * 256 + tid;
        int c = idx >> 4, m = idx & 15;
        float v = feat_l[(size_t)(b * IN_NC + c) * HWc + ypos * Wc + x0 + m];
        xs[m * XS_S + c] = v * (1.0f + atten[b * IN_NC + c]);
    }
    // ---- stage feat_up half of z: zs[m][128+c] = 2 * fup ----
    #pragma unroll
    for (int i = 0; i < 8; ++i) {
        int idx = i * 256 + tid;
        int m = idx >> 7, c = idx & 127;
        int g = c >> 4, cg = c & 15;
        float v = fup[((size_t)(b * Gg + g) * HWc + ypos * Wc + x0 + m) * CG + cg];
        zs[m * ZS_S + 128 + c] = 2.0f * v;
    }
    __syncthreads();

    const int row  = lane & 15;        // A row (pixel) / B,C column
    const int koff = (lane >> 4) * 2;  // k sub-offset for this half-wave
    const int mhi  = (lane >> 4) * 8;  // C row offset

    // ---- GEMM1: feat_arm[16x128] = xs @ fsm_conv_w^T (wave n -> 16 out ch) ----
    {
        const int n0 = wave * 16;
        const float* wrow = fsm_conv_w + (size_t)(n0 + row) * IN_NC;
        v8f c = {};
        #pragma unroll
        for (int k0 = 0; k0 < IN_NC; k0 += 4) {
            v2f a, bb;
            a.x  = xs[row * XS_S + k0 + koff];
            a.y  = xs[row * XS_S + k0 + koff + 1];
            bb.x = wrow[k0 + koff];
            bb.y = wrow[k0 + koff + 1];
            c = __builtin_amdgcn_wmma_f32_16x16x4_f32(false, a, false, bb,
                                                      (short)0, c, false, false);
        }
        #pragma unroll
        for (int r = 0; r < 8; ++r) zs[(r + mhi) * ZS_S + n0 + row] = c[r];
    }
    __syncthreads();

    // ---- GEMM2: om[16x216] = zs @ om_w^T + om_b (14 padded N-tiles over 8 waves) ----
    for (int nt = wave; nt < 14; nt += 8) {
        const int oc = nt * 16 + row;
        const bool okc = oc < OMCH;
        const float* wrow = om_w + (size_t)(okc ? oc : 0) * ZCH;
        v8f c = {};
        #pragma unroll 4
        for (int k0 = 0; k0 < ZCH; k0 += 4) {
            v2f a, bb;
            a.x  = zs[row * ZS_S + k0 + koff];
            a.y  = zs[row * ZS_S + k0 + koff + 1];
            bb.x = okc ? wrow[k0 + koff]     : 0.0f;
            bb.y = okc ? wrow[k0 + koff + 1] : 0.0f;
            c = __builtin_amdgcn_wmma_f32_16x16x4_f32(false, a, false, bb,
                                                      (short)0, c, false, false);
        }
        const float bias = okc ? om_b[oc] : 0.0f;
        #pragma unroll
        for (int r = 0; r < 8; ++r)
            if (okc) oms[(r + mhi) * OM_S + oc] = c[r] + bias;
    }
    __syncthreads();

    // ---- GEMM3: DCNv2 for group g = wave; A (16px x 144) generated on the fly ----
    {
        const int g = wave;
        const float* fupg = fup + (size_t)(b * Gg + g) * HWc * CG;
        const float* wB   = dcn_w + (size_t)(g * CG + row) * (CG * 9); // B col n = row
        v8f c = {};
        #pragma unroll
        for (int t = 0; t < 9; ++t) {
            // sampling geometry for (pixel=row, tap=t) — shared by the 4 ci sub-steps
            float dy  = oms[row * OM_S + g * 18 + t * 2 + 0];
            float dx  = oms[row * OM_S + g * 18 + t * 2 + 1];
            float msk = sigmoidf(oms[row * OM_S + 144 + g * 9 + t]);
            float sy = dy + (float)(t / 3 - 1) + (float)ypos;
            float sx = dx + (float)(t % 3 - 1) + (float)(x0 + row);
            float fy = floorf(sy), fx = floorf(sx);
            float wy = sy - fy,    wx = sx - fx;
            int iy0 = (int)fy, ix0 = (int)fx;
            int iy1 = iy0 + 1, ix1 = ix0 + 1;
            float vy0 = (iy0 >= 0 && iy0 < Hc) ? 1.0f : 0.0f;
            float vy1 = (iy1 >= 0 && iy1 < Hc) ? 1.0f : 0.0f;
            float vx0 = (ix0 >= 0 && ix0 < Wc) ? 1.0f : 0.0f;
            float vx1 = (ix1 >= 0 && ix1 < Wc) ? 1.0f : 0.0f;
            int cy0 = imin(imax(iy0, 0), Hc - 1), cy1 = imin(imax(iy1, 0), Hc - 1);
            int cx0 = imin(imax(ix0, 0), Wc - 1), cx1 = imin(imax(ix1, 0), Wc - 1);
            const float* p00 = fupg + (cy0 * Wc + cx0) * CG;   // 64B-contiguous corners
            const float* p01 = fupg + (cy0 * Wc + cx1) * CG;
            const float* p10 = fupg + (cy1 * Wc + cx0) * CG;
            const float* p11 = fupg + (cy1 * Wc + cx1) * CG;
            float w00 = msk * (1.0f - wy) * (1.0f - wx) * vy0 * vx0;
            float w01 = msk * (1.0f - wy) * wx          * vy0 * vx1;
            float w10 = msk * wy          * (1.0f - wx) * vy1 * vx0;
            float w11 = msk * wy          * wx          * vy1 * vx1;
            #pragma unroll
            for (int ci0 = 0; ci0 < CG; ci0 += 4) {
                const int ci = ci0 + koff;
                v2f a, bb;
                a.x = w00 * p00[ci]   + w01 * p01[ci]   + w10 * p10[ci]   + w11 * p11[ci];
                a.y = w00 * p00[ci+1] + w01 * p01[ci+1] + w10 * p10[ci+1] + w11 * p11[ci+1];
                bb.x = wB[ci * 9 + t];        // dcn_w[((g*16+row)*16 + ci)*9 + t]
                bb.y = wB[(ci + 1) * 9 + t];
                c = __builtin_amdgcn_wmma_f32_16x16x4_f32(false, a, false, bb,
                                                          (short)0, c, false, false);
            }
        }
        // epilogue: +bias, ReLU, +feat_arm; stage in xs (per-wave columns) for coalesced store
        const float biasv = dcn_b[g * CG + row];
        #pragma unroll
        for (int r = 0; r < 8; ++r) {
            int M = r + mhi;
            float v = c[r] + biasv;
            v = v > 0.0f ? v : 0.0f;
            v += zs[M * ZS_S + g * CG + row];          // feat_arm residual
            xs[M * XS_S + g * CG + row] = v;
        }
        #pragma unroll
        for (int i = 0; i < 8; ++i) {
            int idx = i * 32 + lane;                   // (oc, col) over this wave's 16x16 tile
            int ocl = idx >> 4, col = idx & 15;
            out[(size_t)(b * OUT_NC + g * CG + ocl) * HWc + ypos * Wc + x0 + col] =
                xs[col * XS_S + g * CG + ocl];
        }
    }
}

// ---------------- host-side launcher ----------------
extern "C" void kernel_launch(void* const* d_in, const int* in_sizes, int n_in,
                              void* d_out, int out_size, void* d_ws, size_t ws_size,
                              hipStream_t stream) {
    const float* feat_l      = (const float*)d_in[0];
    const float* feat_s      = (const float*)d_in[1];
    const float* fsm_atten_w = (const float*)d_in[2];
    const float* fsm_conv_w  = (const float*)d_in[3];
    const float* om_w        = (const float*)d_in[4];
    const float* om_b        = (const float*)d_in[5];
    const float* dcn_w       = (const float*)d_in[6];
    const float* dcn_b       = (const float*)d_in[7];
    float* out = (float*)d_out;

    float* ws     = (float*)d_ws;
    float* pooled = ws;                       // B*128
    float* atten  = ws + 512;                 // B*128
    float* fup    = ws + 1024;                // B*G*HW*CG = 8,388,608 floats (~33.5 MB)

    pool_kernel<<<Bc * IN_NC, 256, 0, stream>>>(feat_l, pooled);
    atten_kernel<<<Bc, 128, 0, stream>>>(pooled, fsm_atten_w, atten);

    const int n_up = Bc * Gg * HWc * CG;
    upsample_kernel<<<n_up / 256, 256, 0, stream>>>(feat_s, fup);

    const int n_tiles = Bc * HWc / 16;        // 4096 tiles of 16 pixels
    mega_kernel<<<n_tiles, 256, 0, stream>>>(feat_l, atten, fup, fsm_conv_w,
                                             om_w, om_b, dcn_w, dcn_b, out);
}